// RSEBERTAttention_91250875171397
// MI455X (gfx1250) — compile-verified
//
#include <hip/hip_runtime.h>

typedef float v2f __attribute__((ext_vector_type(2)));
typedef float v8f __attribute__((ext_vector_type(8)));

#define DHEAD 64
#define BN    32           // key block (BLOCK_N in reference)
#define MBLK  64           // queries per workgroup (4 waves x 16 rows)
#define KSTR  68           // LDS stride (floats) for roped-K rows: 16B aligned rows
#define VSTR  36           // LDS stride (floats) for V^T rows: 8B aligned b64 reads
#define PSTR  36           // LDS stride (floats) for P staging
#define KSZ   (BN * KSTR)  // floats per K buffer
#define VTSZ  (DHEAD * VSTR)

// xor-butterfly within 16-lane groups via ds_swizzle (and=0x1F, or=0, xor=MASK)
template <int MASK>
__device__ __forceinline__ float swz(float x) {
    return __int_as_float(
        __builtin_amdgcn_ds_swizzle(__float_as_int(x), (MASK << 10) | 0x1F));
}
__device__ __forceinline__ float rowmax16(float x) {
    x = fmaxf(x, swz<1>(x)); x = fmaxf(x, swz<2>(x));
    x = fmaxf(x, swz<4>(x)); x = fmaxf(x, swz<8>(x));
    return x;
}
__device__ __forceinline__ float rowsum16(float x) {
    x += swz<1>(x); x += swz<2>(x); x += swz<4>(x); x += swz<8>(x);
    return x;
}

__global__ __launch_bounds__(128)
void rse_attn(const float* __restrict__ q, const float* __restrict__ k,
              const float* __restrict__ v, const float* __restrict__ cosc,
              const float* __restrict__ sinc, const float* __restrict__ lamp,
              float* __restrict__ out, int N)
{
    __shared__ float sK [2 * KSZ];        // roped K block, row-major [32][KSTR], x2
    __shared__ float sVT[2 * VTSZ];       // V^T block [64][VSTR], x2
    __shared__ float sP [4 * 16 * PSTR];  // per-wave P staging [16][PSTR]

    const int tid  = threadIdx.x;
    const int lane = tid & 31;
    const int wv   = tid >> 5;
    const int ln16 = lane & 15;
    const int hi   = lane >> 4;            // 0: lanes 0-15, 1: lanes 16-31
    const int m0   = blockIdx.x * MBLK + wv * 16;
    const size_t hbase = (size_t)blockIdx.y * N * DHEAD;

    const float lam      = lamp[0];
    const float qk_scale = 0.125f * 1.44269504089f;   // 1/sqrt(64) * log2(e)

    // ---- Q strip (rows m0..m0+15) into WMMA A chunks, rope+scale applied.
    // A 16x4 f32 layout: lane = row (mod 16); lanes>=16 hold K = 4kk+2,+3.
    v2f qA[16];
    {
        const int    row  = m0 + ln16;
        const float* qrow = q    + hbase + (size_t)row * DHEAD;
        const float* crow = cosc + (size_t)row * (DHEAD / 2);
        const float* srow = sinc + (size_t)row * (DHEAD / 2);
        const int    off  = 2 * hi;
#pragma unroll
        for (int kk = 0; kk < 16; ++kk) {
            const int d0 = 4 * kk + off;               // even
            v2f   qv = *(const v2f*)(qrow + d0);
            float c = crow[d0 >> 1], s = srow[d0 >> 1];
            qA[kk].x = qv.x * (c + s) * qk_scale;      // even d -> +sin
            qA[kk].y = qv.y * (c - s) * qk_scale;      // odd  d -> -sin
        }
    }

    // per-lane state for this lane's 8 rows (m = m0 + r + 8*hi)
    float stick[8], li[8], mi[8];
#pragma unroll
    for (int r = 0; r < 8; ++r) { stick[r] = 1.f; li[r] = 0.f; mi[r] = -INFINITY; }

    v8f acc[4];                                        // 16 x 64 output accum
#pragma unroll
    for (int t = 0; t < 4; ++t)
#pragma unroll
        for (int e = 0; e < 8; ++e) acc[t][e] = 0.f;

    const int NB = N / BN;

    // ---------- fill helpers (expanded inline) ----------
    // async V fill: transpose-scatter V[32x64] -> sVT[64][VSTR] via the gfx1250
    // async-copy engine (no VGPR round trip, tracked by ASYNCcnt)
    auto asyncV = [&](int j, int nb) {
        const unsigned long long vbase =
            (unsigned long long)(v + hbase + (size_t)j * BN * DHEAD);
        const unsigned ldsbase =
            (unsigned)(unsigned long long)&sVT[nb * VTSZ];
#pragma unroll
        for (int i = 0; i < 16; ++i) {
            const int e = i * 128 + tid;               // e = n*64 + d
            const int n = e >> 6, d = e & 63;
            const unsigned voff  = (unsigned)(e * 4);
            const unsigned laddr = ldsbase + (unsigned)((d * VSTR + n) * 4);
            asm volatile("global_load_async_to_lds_b32 %0, %1, %2"
                         :: "v"(laddr), "v"(voff), "s"(vbase) : "memory");
        }
    };

    float4 kreg[4];
    v2f    kc[4], ks2[4];
    auto loadK = [&](int j) {                          // issue global loads early
#pragma unroll
        for (int i = 0; i < 4; ++i) {
            const int e   = i * 128 + tid;             // 512 float4 per 32x64 block
            const int row = e >> 4, c4 = e & 15;
            const int d0  = 4 * c4;
            const int p   = j * BN + row;
            kreg[i] = *(const float4*)(k + hbase + (size_t)p * DHEAD + d0);
            kc[i]   = *(const v2f*)(cosc + (size_t)p * 32 + (d0 >> 1));
            ks2[i]  = *(const v2f*)(sinc + (size_t)p * 32 + (d0 >> 1));
        }
    };
    auto storeK = [&](int nb) {                        // rope + LDS store (late)
#pragma unroll
        for (int i = 0; i < 4; ++i) {
            const int e   = i * 128 + tid;
            const int row = e >> 4, c4 = e & 15;
            const int d0  = 4 * c4;
            float4 kr;
            kr.x = kreg[i].x * (kc[i].x + ks2[i].x);
            kr.y = kreg[i].y * (kc[i].x - ks2[i].x);
            kr.z = kreg[i].z * (kc[i].y + ks2[i].y);
            kr.w = kreg[i].w * (kc[i].y - ks2[i].y);
            *(float4*)(&sK[nb * KSZ + row * KSTR + d0]) = kr;
        }
    };

    // ---------- prologue: fill buffer 0 ----------
    asyncV(0, 0);
    loadK(0);
    storeK(0);
    asm volatile("s_wait_asynccnt 0x0" ::: "memory");
    __syncthreads();

    int buf = 0;
    for (int j = 0; j < NB; ++j) {
        const bool nxt = (j + 1 < NB);
        if (nxt) {                                     // start next-block fills
            asyncV(j + 1, buf ^ 1);
            loadK(j + 1);
        }
        const float* cK  = &sK[buf * KSZ];
        const float* cVT = &sVT[buf * VTSZ];

        // ---- S = Qr @ Kr^T : two 16x16 f32 tiles, K=64 -> 16 WMMA each
        v8f s0 = {0.f,0.f,0.f,0.f,0.f,0.f,0.f,0.f};
        v8f s1 = {0.f,0.f,0.f,0.f,0.f,0.f,0.f,0.f};
        const int boff = 2 * hi;                       // B: lanes>=16 hold K+2,+3
#pragma unroll
        for (int dd = 0; dd < 16; ++dd) {
            v2f b0 = *(const v2f*)(&cK[(ln16)      * KSTR + 4 * dd + boff]);
            v2f b1 = *(const v2f*)(&cK[(ln16 + 16) * KSTR + 4 * dd + boff]);
            s0 = __builtin_amdgcn_wmma_f32_16x16x4_f32(false, qA[dd], false, b0,
                                                       (short)0, s0, false, false);
            s1 = __builtin_amdgcn_wmma_f32_16x16x4_f32(false, qA[dd], false, b1,
                                                       (short)0, s1, false, false);
        }

        // ---- log-sigmoid stick-breaking weights (exact reference math)
        const float pk0 = (float)(j * BN + ln16);      // key pos, tile 0
        const float pk1 = pk0 + 16.f;                  // key pos, tile 1
        float w0[8], w1[8];
#pragma unroll
        for (int r = 0; r < 8; ++r) {
            const float pm = (float)(m0 + r + 8 * hi); // query pos for this row
            const float l0 = s0[r] - lam * (pk0 - pm);
            const float l1 = s1[r] - lam * (pk1 - pm);
            const float b0e = __expf(l0 - __logf(1.f + __expf(fminf(l0, 20.f))));
            const float b1e = __expf(l1 - __logf(1.f + __expf(fminf(l1, 20.f))));
            const float bm = rowmax16(fmaxf(b0e, b1e));          // beta.max
            const float lm = rowmax16(fmaxf(l0, l1));            // logits.max
            float ww0 = b0e * stick[r];
            float ww1 = b1e * stick[r];
            ww0 = (ww0 > 1e-8f) ? ww0 : 0.f;
            ww1 = (ww1 > 1e-8f) ? ww1 : 0.f;
            const float ws  = rowsum16(ww0 + ww1);
            const float inv = __builtin_amdgcn_rcpf(ws + 1e-8f);
            w0[r] = ww0 * inv;                          // normalized weights
            w1[r] = ww1 * inv;
            const float mnew = fmaxf(mi[r], lm);
            li[r]   = li[r] * exp2f(mi[r] - mnew) + ws * inv;
            mi[r]   = mnew;
            stick[r] = stick[r] * fmaxf(1.f - bm, 0.1f);
        }

        // ---- stage P through per-wave LDS: C-layout -> A-layout transpose
        float* pw = &sP[wv * 16 * PSTR];
#pragma unroll
        for (int r = 0; r < 8; ++r) {
            pw[(r + 8 * hi) * PSTR + ln16]      = w0[r];
            pw[(r + 8 * hi) * PSTR + ln16 + 16] = w1[r];
        }
        // DS ops are in-order within a wave: reload as A operand directly
        v2f aP[8];
#pragma unroll
        for (int kk = 0; kk < 8; ++kk)
            aP[kk] = *(const v2f*)(&pw[ln16 * PSTR + 4 * kk + 2 * hi]);

        // ---- acc += P @ V : 4 d-tiles x 8 chunks of K=4
#pragma unroll
        for (int dt = 0; dt < 4; ++dt) {
#pragma unroll
            for (int kk = 0; kk < 8; ++kk) {
                v2f bV = *(const v2f*)(&cVT[(dt * 16 + ln16) * VSTR + 4 * kk + 2 * hi]);
                acc[dt] = __builtin_amdgcn_wmma_f32_16x16x4_f32(false, aP[kk], false, bV,
                                                                (short)0, acc[dt], false, false);
            }
        }

        // ---- finish next-block fills, publish, swap buffers
        if (nxt) storeK(buf ^ 1);
        asm volatile("s_wait_asynccnt 0x0" ::: "memory");
        __syncthreads();
        buf ^= 1;
    }

    // ---- epilogue: out = acc / max(l_i, 1e-6)
    float inv_l[8];
#pragma unroll
    for (int r = 0; r < 8; ++r)
        inv_l[r] = __builtin_amdgcn_rcpf(fmaxf(li[r], 1e-6f));
#pragma unroll
    for (int dt = 0; dt < 4; ++dt)
#pragma unroll
        for (int r = 0; r < 8; ++r)
            out[hbase + (size_t)(m0 + r + 8 * hi) * DHEAD + dt * 16 + ln16] =
                acc[dt][r] * inv_l[r];
}

extern "C" void kernel_launch(void* const* d_in, const int* in_sizes, int n_in,
                              void* d_out, int out_size, void* d_ws, size_t ws_size,
                              hipStream_t stream) {
    (void)n_in; (void)out_size; (void)d_ws; (void)ws_size;
    const float* q    = (const float*)d_in[0];
    const float* k    = (const float*)d_in[1];
    const float* v    = (const float*)d_in[2];
    const float* cosc = (const float*)d_in[3];
    const float* sinc = (const float*)d_in[4];
    const float* lam  = (const float*)d_in[5];
    float* out = (float*)d_out;

    const int Nseq = in_sizes[3] / (DHEAD / 2);        // cos cache is [N, D/2]
    const int BH   = in_sizes[0] / (Nseq * DHEAD);     // B*H
    dim3 grid(Nseq / MBLK, BH);
    rse_attn<<<grid, dim3(128), 0, stream>>>(q, k, v, cosc, sinc, lam, out, Nseq);
}